// QubicAttention_3d_64269890617734
// MI455X (gfx1250) — compile-verified
//
#include <hip/hip_runtime.h>

// Problem constants (from reference): B=2, C=64, D=32, H=64, W=64
#define Cch   64
#define Dd    32
#define Hh    64
#define Ww    64
#define SPAT  (Dd * Hh * Ww)      // 131072 spatial per batch
#define NG    (2 * Dd * Hh)       // 4096 groups of 64 consecutive spatial positions
#define LSTR  68                  // padded LDS row stride (floats)

typedef float v2f __attribute__((ext_vector_type(2)));
typedef float v8f __attribute__((ext_vector_type(8)));

static __device__ inline v8f vsplat8(float v) {
  v8f r;
#pragma unroll
  for (int j = 0; j < 8; ++j) r[j] = v;
  return r;
}

static __device__ inline v8f wmma4(v2f a, v2f b, v8f c) {
  // D = A(16x4 f32) * B(4x16 f32) + C(16x16 f32)
  return __builtin_amdgcn_wmma_f32_16x16x4_f32(false, a, false, b, (short)0, c,
                                               false, false);
}

// A stored [m][k] (k contiguous, stride astr); B stored transposed: Bt[n][k]
// (k contiguous, stride bstr).  Computes 16x16 tile D(m0,n0) += A*B.
template <int K>
static __device__ inline v8f mm16_nt(const float* A, int astr, const float* Bt,
                                     int bstr, int m0, int n0, v8f acc) {
  const int lane = threadIdx.x & 31;
  const int r    = lane & 15;
  const int ko   = (lane >> 4) << 1;  // 0 for lanes 0-15, 2 for lanes 16-31
  const float* ap = A  + (m0 + r) * astr + ko;
  const float* bp = Bt + (n0 + r) * bstr + ko;
#pragma unroll
  for (int k = 0; k < K; k += 4) {
    v2f av = *(const v2f*)(ap + k);
    v2f bv = *(const v2f*)(bp + k);
    acc = wmma4(av, bv, acc);
  }
  return acc;
}

// A stored [m][k]; B stored natural: Bkn[k][n] (n contiguous, stride bstr).
template <int K>
static __device__ inline v8f mm16_nn(const float* A, int astr, const float* Bkn,
                                     int bstr, int m0, int n0, v8f acc) {
  const int lane = threadIdx.x & 31;
  const int r    = lane & 15;
  const int ko   = (lane >> 4) << 1;
  const float* ap = A + (m0 + r) * astr + ko;
  const float* bp = Bkn + ko * bstr + n0 + r;
#pragma unroll
  for (int k = 0; k < K; k += 4) {
    v2f av = *(const v2f*)(ap + k);
    v2f bv;
    bv[0] = bp[k * bstr];
    bv[1] = bp[k * bstr + bstr];
    acc = wmma4(av, bv, acc);
  }
  return acc;
}

// Store 16x16 f32 D-tile (ISA layout: lanes 0-15 rows M=j, lanes 16-31 M=j+8)
static __device__ inline void st_tile(float* dst, int dstr, int m0, int n0, v8f acc) {
  const int lane = threadIdx.x & 31;
  const int n  = n0 + (lane & 15);
  const int mb = m0 + ((lane >> 4) << 3);
#pragma unroll
  for (int j = 0; j < 8; ++j) dst[(mb + j) * dstr + n] = acc[j];
}

static __device__ inline void softmax_row(float* row, int L) {
  float mx = row[0];
  for (int j = 1; j < L; ++j) mx = fmaxf(mx, row[j]);
  float s = 0.f;
  for (int j = 0; j < L; ++j) {
    float e = __expf(row[j] - mx);
    row[j] = e;
    s += e;
  }
  float inv = 1.f / s;
  for (int j = 0; j < L; ++j) row[j] *= inv;
}

// ---------------- kernel 1: short = conv1x1(x, ws) + bs  (plain stores) -----
__global__ __launch_bounds__(256) void k_short(const float* __restrict__ x,
                                               const float* __restrict__ ws,
                                               const float* __restrict__ bs,
                                               float* __restrict__ out) {
  __shared__ __align__(16) float xT[64 * LSTR];  // [w][c]
  __shared__ __align__(16) float Wt[64 * LSTR];  // [o][c]
  const int tid = threadIdx.x, wave = tid >> 5, lane = tid & 31;
  const int g = blockIdx.x;
  const int b = g >> 11, d = (g >> 6) & 31, h = g & 63;
  const size_t xoff = (size_t)b * Cch * SPAT + (size_t)((d * Hh + h) * Ww);
  for (int idx = tid; idx < 4096; idx += 256) {
    const int c = idx >> 6, w = idx & 63;
    xT[w * LSTR + c] = x[xoff + (size_t)c * SPAT + w];
    Wt[c * LSTR + w] = ws[idx];  // Wt[idx>>6][idx&63]
  }
  __syncthreads();
  const size_t obase = (size_t)b * Cch * SPAT + (size_t)((d * Hh + h) * Ww);
#pragma unroll
  for (int q = 0; q < 2; ++q) {
    const int t = wave * 2 + q;
    const int m0 = (t >> 2) << 4, n0 = (t & 3) << 4;  // m = w, n = c_out
    v8f acc = vsplat8(bs[n0 + (lane & 15)]);
    acc = mm16_nt<64>(xT, LSTR, Wt, LSTR, m0, n0, acc);
    const int n  = n0 + (lane & 15);
    const int mb = m0 + ((lane >> 4) << 3);
#pragma unroll
    for (int j = 0; j < 8; ++j)
      out[obase + (size_t)n * SPAT + (mb + j)] = acc[j];
  }
}

// ---------------- kernel 2: one axial attention pass, out += result --------
template <int AXIS>  // 0 = t-axis (L=32, 2 chunks), 1 = h-axis, 2 = w-axis
__global__ __launch_bounds__(256) void k_axis(const float* __restrict__ x,
                                              const float* __restrict__ wq,
                                              const float* __restrict__ bq,
                                              const float* __restrict__ wk,
                                              const float* __restrict__ bk,
                                              const float* __restrict__ wv,
                                              const float* __restrict__ bv,
                                              float* __restrict__ out) {
  __shared__ __align__(16) float xT[64 * LSTR];  // [w][c]
  __shared__ __align__(16) float WS[64 * LSTR];  // weights, then scores
  __shared__ __align__(16) float Qb[64 * LSTR];  // [w][c]
  __shared__ __align__(16) float Kb[64 * LSTR];
  __shared__ __align__(16) float Vb[64 * LSTR];
  const int tid = threadIdx.x, wave = tid >> 5, lane = tid & 31;
  const int g = blockIdx.x;
  const int b = g >> 11, d = (g >> 6) & 31, h = g & 63;
  const size_t xoff = (size_t)b * Cch * SPAT + (size_t)((d * Hh + h) * Ww);
  for (int idx = tid; idx < 4096; idx += 256)
    xT[(idx & 63) * LSTR + (idx >> 6)] = x[xoff + (size_t)(idx >> 6) * SPAT + (idx & 63)];

  // ---- projections Q,K,V for this axis ----
  for (int p = 0; p < 3; ++p) {
    const float* wm = (p == 0 ? wq : p == 1 ? wk : wv) + AXIS * 4096;
    const float* bb = (p == 0 ? bq : p == 1 ? bk : bv) + AXIS * 64;
    float* dst = (p == 0 ? (float*)Qb : p == 1 ? (float*)Kb : (float*)Vb);
    __syncthreads();  // previous WS use done
    for (int idx = tid; idx < 4096; idx += 256)
      WS[(idx >> 6) * LSTR + (idx & 63)] = wm[idx];
    __syncthreads();
#pragma unroll
    for (int q = 0; q < 2; ++q) {
      const int t = wave * 2 + q;
      const int m0 = (t >> 2) << 4, n0 = (t & 3) << 4;
      v8f acc = vsplat8(bb[n0 + (lane & 15)]);
      acc = mm16_nt<64>(xT, LSTR, WS, LSTR, m0, n0, acc);
      st_tile(dst, LSTR, m0, n0, acc);
    }
  }
  __syncthreads();

  const size_t ob = (size_t)b * Cch * SPAT;

  if constexpr (AXIS == 0) {
    // t-axis: two L=32 chunks (w in [0,32), [32,64)).  S = K Q^T per chunk.
    {
      const int chunk = wave >> 2, st = wave & 3;
      const int m0 = chunk * 32 + ((st >> 1) << 4);
      const int n0 = chunk * 32 + ((st & 1) << 4);
      v8f acc = vsplat8(0.f);
      acc = mm16_nt<64>(Kb, LSTR, Qb, LSTR, m0, n0, acc);
      st_tile(WS, LSTR, m0, (st & 1) << 4, acc);
    }
    __syncthreads();
    if (tid < 64) softmax_row(WS + tid * LSTR, 32);
    __syncthreads();
#pragma unroll
    for (int q = 0; q < 2; ++q) {
      const int t = wave * 2 + q;
      const int chunk = t >> 3, r = t & 7;
      const int mloc = (r >> 2) << 4, n0 = (r & 3) << 4;
      v8f acc = vsplat8(0.f);
      acc = mm16_nn<32>(WS + chunk * 32 * LSTR, LSTR, Vb + chunk * 32 * LSTR,
                        LSTR, mloc, n0, acc);
      const int n  = n0 + (lane & 15);
      const int mb = mloc + ((lane >> 4) << 3);
      const int hh  = 2 * h + chunk;            // chunk id bits within (d,h)
      const int h_o = 2 * d + (hh >> 6);
      const int w_o = hh & 63;
      const size_t base = ob + (size_t)n * SPAT + (size_t)(h_o * Ww + w_o);
#pragma unroll
      for (int j = 0; j < 8; ++j) {
        const int i = mb + j;  // row within chunk -> output d index
        out[base + (size_t)i * (Hh * Ww)] += acc[j];
      }
    }
  } else {
    // h/w axes: single L=64 chunk over this (b,d,h) row of w.
#pragma unroll
    for (int q = 0; q < 2; ++q) {
      const int t = wave * 2 + q;
      const int m0 = (t >> 2) << 4, n0 = (t & 3) << 4;
      v8f acc = vsplat8(0.f);
      acc = mm16_nt<64>(Kb, LSTR, Qb, LSTR, m0, n0, acc);
      st_tile(WS, LSTR, m0, n0, acc);
    }
    __syncthreads();
    if (tid < 64) softmax_row(WS + tid * LSTR, 64);
    __syncthreads();
#pragma unroll
    for (int q = 0; q < 2; ++q) {
      const int t = wave * 2 + q;
      const int m0 = (t >> 2) << 4, n0 = (t & 3) << 4;
      v8f acc = vsplat8(0.f);
      acc = mm16_nn<64>(WS, LSTR, Vb, LSTR, m0, n0, acc);
      const int n  = n0 + (lane & 15);
      const int mb = m0 + ((lane >> 4) << 3);
#pragma unroll
      for (int j = 0; j < 8; ++j) {
        const int m = mb + j;  // attention row index (= source w position)
        size_t idx;
        if constexpr (AXIS == 2)
          idx = ob + (size_t)n * SPAT + (size_t)((d * Hh + h) * Ww + m); // direct
        else
          idx = ob + (size_t)n * SPAT + (size_t)((d * Hh + m) * Ww + h); // h<->w
        out[idx] += acc[j];
      }
    }
  }
}

extern "C" void kernel_launch(void* const* d_in, const int* in_sizes, int n_in,
                              void* d_out, int out_size, void* d_ws,
                              size_t ws_size, hipStream_t stream) {
  (void)in_sizes; (void)n_in; (void)out_size; (void)d_ws; (void)ws_size;
  const float* x  = (const float*)d_in[0];
  const float* wq = (const float*)d_in[1];
  const float* bq = (const float*)d_in[2];
  const float* wk = (const float*)d_in[3];
  const float* bk = (const float*)d_in[4];
  const float* wv = (const float*)d_in[5];
  const float* bv = (const float*)d_in[6];
  const float* ws = (const float*)d_in[7];
  const float* bs = (const float*)d_in[8];
  float* out = (float*)d_out;

  k_short<<<NG, 256, 0, stream>>>(x, ws, bs, out);                 // out = short
  k_axis<0><<<NG, 256, 0, stream>>>(x, wq, bq, wk, bk, wv, bv, out); // += out_t
  k_axis<1><<<NG, 256, 0, stream>>>(x, wq, bq, wk, bk, wv, bv, out); // += out_h
  k_axis<2><<<NG, 256, 0, stream>>>(x, wq, bq, wk, bk, wv, bv, out); // += out_v
}